// ThermoPBMLayer_62603443306678
// MI455X (gfx1250) — compile-verified
//
#include <hip/hip_runtime.h>
#include <hip/hip_bf16.h>

typedef float v2f __attribute__((ext_vector_type(2)));
typedef float v8f __attribute__((ext_vector_type(8)));
typedef unsigned int v4u __attribute__((ext_vector_type(4)));
typedef int v8i __attribute__((ext_vector_type(8)));
typedef int v4i __attribute__((ext_vector_type(4)));

#define NBATCH 4096
#define NROOM  64
#define WAVES_PER_BLOCK 8
#define LDS_STRIDE 65   // 64 data dwords + 1 pad dword inserted by TDM padding

// Fast reciprocal: v_rcp_f32 (~1 ulp) + one Newton-Raphson step (~0.5 ulp).
// Branchless; callers mask the den==0 case per divide_no_nan semantics.
__device__ __forceinline__ float fast_rcp(float d) {
    float r = __builtin_amdgcn_rcpf(d);
    float e = __builtin_fmaf(-d, r, 1.0f);
    return __builtin_fmaf(r, e, r);
}

// inverse_lin_comb with divide_no_nan semantics:
//   on*off / (on*(1-x) + off*x), 0 when denominator == 0
__device__ __forceinline__ float inv_lin_comb(float on, float off, float x) {
    float den = __builtin_fmaf(off - on, x, on);   // on*(1-x) + off*x
    float r   = (on * off) * fast_rcp(den);
    return (den == 0.0f) ? 0.0f : r;               // single v_cndmask, no branch
}

__device__ __forceinline__ float div_no_nan(float a, float b) {
    float r = a * fast_rcp(b);
    return (b == 0.0f) ? 0.0f : r;
}

// lin_comb(on, off, x) = on*x + off*(1-x)
__device__ __forceinline__ float lin_comb(float on, float off, float x) {
    return __builtin_fmaf(on - off, x, off);
}

// Issue one TDM descriptor: DMA a 64x64 fp32 matrix (contiguous in global) into
// LDS at lds_byte, inserting 1 pad DWORD after every 64 DWORDs (row stride 65).
__device__ __forceinline__ void tdm_load_64x64_padded(const float* gptr, unsigned lds_byte) {
    unsigned long long ga = (unsigned long long)(size_t)gptr;
    v4u g0;
    g0.x = 1u;                                     // count=1 (valid descriptor)
    g0.y = lds_byte;                               // lds_addr
    g0.z = (unsigned)(ga & 0xFFFFFFFFull);         // global_addr[31:0]
    g0.w = (unsigned)((ga >> 32) & 0x01FFFFFFull)  // global_addr[56:32]
         | 0x80000000u;                            // type=2 ("image")
    v8i g1;
    g1[0] = (int)((2u << 16)      // data_size = 2 -> 4 bytes
                | (1u << 20)      // pad_enable
                | (5u << 22));    // pad_interval = 5 -> every 64 DWORDs (pad_amount=0 -> 1 DWORD)
    g1[1] = (int)(64u << 16);     // tensor_dim0 = 64 (low 16 bits at [31:16])
    g1[2] = (int)(64u << 16);     // tensor_dim1 = 64
    g1[3] = (int)(64u << 16);     // tile_dim0 = 64
    g1[4] = 64;                   // tile_dim1 = 64 (tile_dim2 = 0)
    g1[5] = 64;                   // tensor_dim0_stride = 64
    g1[6] = 0;
    g1[7] = 0;
    v4i z4 = {0, 0, 0, 0};             // 2D tensor: groups 2/3 unused
    v8i z8 = {0, 0, 0, 0, 0, 0, 0, 0}; // trailing group (clang-23 6-arg form)
    __builtin_amdgcn_tensor_load_to_lds(g0, g1, z4, z4, z8, 0);
}

__global__ __launch_bounds__(256) void thermo_pbm_kernel(
    const float* __restrict__ T_rooms,      // (B, R, 1)
    const float* __restrict__ T_walls,      // (B, R, 1)
    const float* __restrict__ T_out,        // (B, 1, 1)
    const float* __restrict__ T_direct,     // (NDC, B, R, 1)
    const float* __restrict__ u_is_on,      // (B, R, NU)
    const float* __restrict__ R_is_on,      // (B, 1, 1)
    const float* __restrict__ direct_is_on, // (NDC, B, R, 1)
    const float* __restrict__ R_int_on,     // (1, R, R)
    const float* __restrict__ R_int_off,    // (1, R, R)
    const float* __restrict__ R_inv_owo,    // (1, R, 1)
    const float* __restrict__ R_inv_ow,     // (1, R, 1)
    const float* __restrict__ R_inv_iw,     // (1, R, 1)
    const float* __restrict__ R_inv_rw,     // (1, R, 1)
    const float* __restrict__ u_dir_on,     // (1, R, NU)
    const float* __restrict__ u_dir_off,
    const float* __restrict__ u_in_on,
    const float* __restrict__ u_in_off,
    const float* __restrict__ u_out_on,
    const float* __restrict__ u_out_off,
    const float* __restrict__ R_dir_on,     // (NDC, 1, R, 1)
    const float* __restrict__ R_dir_off,
    const float* __restrict__ C_inv_rooms,  // (1, R, 1)
    const float* __restrict__ C_inv_walls,  // (1, R, 1)
    float* __restrict__ out)                // (B, R, 2)
{
    __shared__ float s_on [NROOM * LDS_STRIDE];
    __shared__ float s_off[NROOM * LDS_STRIDE];
    __shared__ float s_tr [WAVES_PER_BLOCK][NROOM];
    __shared__ float s_int[WAVES_PER_BLOCK][NROOM];

    const int tid  = threadIdx.x;
    const int wave = tid >> 5;
    const int lane = tid & 31;
    const int b    = blockIdx.x * WAVES_PER_BLOCK + wave;

    // --- Phase 1: TDM DMA of the shared 64x64 mixing matrices into LDS (wave 0) ---
    if (tid < 32) {
        tdm_load_64x64_padded(R_int_on,  (unsigned)(size_t)(void*)s_on);
        tdm_load_64x64_padded(R_int_off, (unsigned)(size_t)(void*)s_off);
        __builtin_amdgcn_s_wait_tensorcnt(0);
    }

    // --- Phase 1b: stage this wave's Tr vector into LDS ---
    s_tr[wave][lane]      = T_rooms[b * NROOM + lane];
    s_tr[wave][lane + 32] = T_rooms[b * NROOM + lane + 32];
    __syncthreads();

    const float x = R_is_on[b];
    const float* trw = s_tr[wave];

    // --- Phase 2: per-batch 64x64 mat-vec via V_WMMA_F32_16X16X4_F32 ---
    // A-fragment slot (lane, vgpr) holds A[m, k] with m = lane&15, k = 2*(lane>>4)+vgpr.
    // B-fragment columns are identical (Tr replicated) so every D column holds y.
    // Second accumulator with B = 1 gives row sums.
    const int m     = lane & 15;
    const int khalf = (lane >> 4) * 2;
    v2f ones; ones.x = 1.0f; ones.y = 1.0f;

    for (int t = 0; t < 4; ++t) {
        const int i0  = t * 16;
        const int row = i0 + m;
        v8f acc  = {0.f, 0.f, 0.f, 0.f, 0.f, 0.f, 0.f, 0.f};
        v8f rsum = {0.f, 0.f, 0.f, 0.f, 0.f, 0.f, 0.f, 0.f};
        #pragma unroll
        for (int kt = 0; kt < 16; ++kt) {
            const int kg = kt * 4 + khalf;
            const int base = row * LDS_STRIDE + kg;
            float on0  = s_on[base],     off0 = s_off[base];
            float on1  = s_on[base + 1], off1 = s_off[base + 1];
            v2f a;  a.x = inv_lin_comb(on0, off0, x);
                    a.y = inv_lin_comb(on1, off1, x);
            v2f bv; bv.x = trw[kg];
                    bv.y = trw[kg + 1];
            acc  = __builtin_amdgcn_wmma_f32_16x16x4_f32(
                       false, a, false, bv,   (short)0, acc,  false, false);
            rsum = __builtin_amdgcn_wmma_f32_16x16x4_f32(
                       false, a, false, ones, (short)0, rsum, false, false);
        }
        // D layout: VGPR r, lanes 0-15 -> M=r ; lanes 16-31 -> M=8+r (all N identical)
        if (m == 0) {
            const int base = i0 + (lane >> 4) * 8;
            #pragma unroll
            for (int r = 0; r < 8; ++r) {
                // internal[i] = sum_j M[i,j]*Tr[j] - Tr[i]*rowsum[i]
                s_int[wave][base + r] = acc[r] - trw[base + r] * rsum[r];
            }
        }
    }
    __syncthreads();

    // --- Phase 3: fused per-(b,i) elementwise pass (each lane handles 2 rows) ---
    const float Tout_b = T_out[b];
    #pragma unroll
    for (int rep = 0; rep < 2; ++rep) {
        const int i  = lane + rep * 32;
        const int gi = b * NROOM + i;
        const float Tr_i = trw[i];
        const float Tw_i = T_walls[gi];

        const float4 u     = *(const float4*)(u_is_on  + (size_t)gi * 4);
        const float4 udon  = *(const float4*)(u_dir_on + i * 4);
        const float4 udoff = *(const float4*)(u_dir_off+ i * 4);
        const float4 uion  = *(const float4*)(u_in_on  + i * 4);
        const float4 uioff = *(const float4*)(u_in_off + i * 4);
        const float4 uoon  = *(const float4*)(u_out_on + i * 4);
        const float4 uooff = *(const float4*)(u_out_off+ i * 4);

        const float uds = lin_comb(udon.x, udoff.x, u.x) + lin_comb(udon.y, udoff.y, u.y)
                        + lin_comb(udon.z, udoff.z, u.z) + lin_comb(udon.w, udoff.w, u.w);
        const float uis = lin_comb(uion.x, uioff.x, u.x) + lin_comb(uion.y, uioff.y, u.y)
                        + lin_comb(uion.z, uioff.z, u.z) + lin_comb(uion.w, uioff.w, u.w);
        const float uos = lin_comb(uoon.x, uooff.x, u.x) + lin_comb(uoon.y, uooff.y, u.y)
                        + lin_comb(uoon.z, uooff.z, u.z) + lin_comb(uoon.w, uooff.w, u.w);

        // direct channels (NDC = 2)
        float dsum = 0.0f;
        #pragma unroll
        for (int c = 0; c < 2; ++c) {
            const int di = c * (NBATCH * NROOM) + gi;
            const float don  = R_dir_on [c * NROOM + i];
            const float doff = R_dir_off[c * NROOM + i];
            const float dx   = direct_is_on[di];
            const float rinv = inv_lin_comb(don, doff, dx);
            dsum += rinv * (T_direct[di] - Tr_i);
        }

        // inner wall
        const float Riw = R_inv_iw[i], Rrw = R_inv_rw[i];
        const float prod = Riw * Rrw * (Tr_i - Tw_i);
        const float rs2  = Riw + Rrw;
        const float rrs2 = fast_rcp(rs2);
        const float to_wall = (rs2 == 0.0f) ? 0.0f : (Riw * uis + prod) * rrs2;
        const float to_room = (rs2 == 0.0f) ? 0.0f : (Rrw * uis - prod) * rrs2;

        // outer wall
        const float Rowo = R_inv_owo[i], Row = R_inv_ow[i];
        const float oww = div_no_nan(Row * uos + (Row * Rowo) * (Tout_b - Tw_i), Row + Rowo);

        const float rhs_rooms = uds + s_int[wave][i] + to_room + dsum;
        const float rhs_walls = to_wall + oww;

        float2 o;
        o.x = Tr_i + 60.0f * (C_inv_rooms[i] * rhs_rooms);
        o.y = Tw_i + 60.0f * (C_inv_walls[i] * rhs_walls);
        *(float2*)(out + (size_t)gi * 2) = o;
    }
}

extern "C" void kernel_launch(void* const* d_in, const int* in_sizes, int n_in,
                              void* d_out, int out_size, void* d_ws, size_t ws_size,
                              hipStream_t stream) {
    (void)in_sizes; (void)n_in; (void)out_size; (void)d_ws; (void)ws_size;
    const float* T_rooms      = (const float*)d_in[0];
    const float* T_walls      = (const float*)d_in[1];
    const float* T_out        = (const float*)d_in[2];
    const float* T_direct     = (const float*)d_in[3];
    const float* u_is_on      = (const float*)d_in[4];
    const float* R_is_on      = (const float*)d_in[5];
    const float* direct_is_on = (const float*)d_in[6];
    const float* R_int_on     = (const float*)d_in[7];
    const float* R_int_off    = (const float*)d_in[8];
    const float* R_inv_owo    = (const float*)d_in[9];
    const float* R_inv_ow     = (const float*)d_in[10];
    const float* R_inv_iw     = (const float*)d_in[11];
    const float* R_inv_rw     = (const float*)d_in[12];
    const float* u_dir_on     = (const float*)d_in[13];
    const float* u_dir_off    = (const float*)d_in[14];
    const float* u_in_on      = (const float*)d_in[15];
    const float* u_in_off     = (const float*)d_in[16];
    const float* u_out_on     = (const float*)d_in[17];
    const float* u_out_off    = (const float*)d_in[18];
    const float* R_dir_on     = (const float*)d_in[19];
    const float* R_dir_off    = (const float*)d_in[20];
    const float* C_inv_rooms  = (const float*)d_in[21];
    const float* C_inv_walls  = (const float*)d_in[22];

    dim3 grid(NBATCH / WAVES_PER_BLOCK);  // 512 blocks, 1 batch per wave32
    dim3 block(32 * WAVES_PER_BLOCK);     // 256 threads = 8 waves
    thermo_pbm_kernel<<<grid, block, 0, stream>>>(
        T_rooms, T_walls, T_out, T_direct, u_is_on, R_is_on, direct_is_on,
        R_int_on, R_int_off, R_inv_owo, R_inv_ow, R_inv_iw, R_inv_rw,
        u_dir_on, u_dir_off, u_in_on, u_in_off, u_out_on, u_out_off,
        R_dir_on, R_dir_off, C_inv_rooms, C_inv_walls, (float*)d_out);
}